// PointGNN_81088982548750
// MI455X (gfx1250) — compile-verified
//
#include <hip/hip_runtime.h>
#include <hip/hip_bf16.h>
#include <math.h>

typedef float v2f __attribute__((ext_vector_type(2)));
typedef float v8f __attribute__((ext_vector_type(8)));

#define NNODES 50000
#define NEDGES 800000
#define NGRAPH 32
#define DF     300
#define NCLS   40
#define KPAD   304   // max padded K (303 edge / 300 node -> 304)

// ---------------------------------------------------------------------------
// float atomic max via signed/unsigned integer atomics (target init = -inf)
// ---------------------------------------------------------------------------
__device__ __forceinline__ void atomicMaxF(float* addr, float val) {
  if (val >= 0.0f) {
    atomicMax((int*)addr, __float_as_int(val));
  } else {
    atomicMin((unsigned int*)addr, __float_as_uint(val));
  }
}

// ---------------------------------------------------------------------------
// Pack B[K,N] (row-major) into WMMA-native padded layout:
//   Bp pair index  p = (c*NPAD + n)*2 + hi   (v2f units)
//   Bp[p] = ( B[4c+2hi+0][n], B[4c+2hi+1][n] )   zero outside K x N
// Each lane's per-chunk B operand is then ONE aligned 8-byte load with
// constant stride NPAD*16 bytes -> branch-free, coalesced hot loop.
// ---------------------------------------------------------------------------
__global__ void pack_b(const float* __restrict__ B, float* __restrict__ Bp,
                       int K, int N, int NPAD, int KC)
{
  long total = (long)KC * NPAD * 4;
  long i = (long)blockIdx.x * blockDim.x + threadIdx.x;
  if (i >= total) return;
  int j  = (int)(i & 1);
  int hi = (int)((i >> 1) & 1);
  long cn = i >> 2;
  int n  = (int)(cn % NPAD);
  int c  = (int)(cn / NPAD);
  int k  = 4 * c + 2 * hi + j;
  float v = (k < K && n < N) ? B[(size_t)k * N + n] : 0.0f;
  Bp[i] = v;
}

// ---------------------------------------------------------------------------
// 16-row-tile WMMA GEMM:  C[M,N] = act(A[M,K] @ B + bias + resid)
// B supplied pre-packed (pack_b).  flags bit0 = relu, bit1 = -inf -> 0 on A.
// block = 128 threads (4 waves); one 16-row m-tile/block staged in LDS;
// waves split the n-tiles.  Hot loop: 1 ds_load_b64 + 1 global b64 + 1 WMMA.
// ---------------------------------------------------------------------------
__global__ __launch_bounds__(128)
void gemm16_wmma(const float* __restrict__ A, const v2f* __restrict__ Bp,
                 const float* __restrict__ bias, const float* __restrict__ resid,
                 float* __restrict__ C, int M, int N, int K,
                 int NPAD, int KC, int flags)
{
  __shared__ float As[16 * KPAD];
  const int m0  = blockIdx.x * 16;
  const int tid = threadIdx.x;

  for (int idx = tid; idx < 16 * KPAD; idx += 128) {
    int m = idx / KPAD;
    int k = idx - m * KPAD;
    int gm = m0 + m;
    float v = 0.0f;
    if (gm < M && k < K) {
      v = A[(size_t)gm * K + k];
      if (flags & 2) {
        if (__float_as_uint(v) == 0xFF800000u) v = 0.0f;  // -inf -> 0
      }
    }
    As[idx] = v;
  }
  __syncthreads();

  const int lane = tid & 31;
  const int wave = tid >> 5;
  const int ln   = lane & 15;
  const int hi   = lane >> 4;             // lane half
  const int koff = hi << 1;               // 0 or 2
  const int mofs = hi << 3;               // 0 or 8
  const int ntiles = NPAD >> 4;
  const long bstride = (long)NPAD * 2;    // v2f units per chunk

  for (int ntile = wave; ntile < ntiles; ntile += 4) {
    const int n = ntile * 16 + ln;
    const v2f* ap = (const v2f*)(As + ln * KPAD + koff);  // +2 per chunk
    const v2f* bp = Bp + ((long)n * 2 + hi);              // +bstride per chunk
    v8f acc = {};
#pragma unroll 4
    for (int c = 0; c < KC; ++c) {
      v2f a = ap[2 * c];
      v2f b = bp[(long)c * bstride];
      acc = __builtin_amdgcn_wmma_f32_16x16x4_f32(false, a, false, b,
                                                  (short)0, acc, false, false);
    }
    if (n < N) {
      const float bv = bias ? bias[n] : 0.0f;
#pragma unroll
      for (int r = 0; r < 8; ++r) {
        const int gm = m0 + mofs + r;
        if (gm < M) {
          float v = acc[r] + bv;
          if (resid) v += resid[(size_t)gm * N + n];
          if (flags & 1) v = v > 0.0f ? v : 0.0f;
          C[(size_t)gm * N + n] = v;
        }
      }
    }
  }
}

// ---------------------------------------------------------------------------
// Fused PointGNN edge kernel:
//   per edge: v = [pos[src]+q[dst] (3) | h[src] (300)]  (K=303 -> 304)
//   e_out = v @ fW + fb  (WMMA f32), scatter-max into agg[dst]
// fW supplied pre-packed (NPAD=304, KC=76).  One 16-edge tile per block.
// ---------------------------------------------------------------------------
__global__ __launch_bounds__(128)
void edge_wmma(const float* __restrict__ h, const float* __restrict__ pos,
               const float* __restrict__ q, const int* __restrict__ src,
               const int* __restrict__ dst, const v2f* __restrict__ fWp,
               const float* __restrict__ fb, float* __restrict__ agg, int E)
{
  __shared__ float As[16 * KPAD];
  __shared__ int   dsts[16];
  const int e0  = blockIdx.x * 16;
  const int tid = threadIdx.x;

  for (int idx = tid; idx < 16 * KPAD; idx += 128) {
    int m = idx / KPAD;
    int k = idx - m * KPAD;
    int e = e0 + m;
    float v = 0.0f;
    if (e < E) {
      int s = src[e];
      if (k < 3) {
        int d = dst[e];
        v = pos[s * 3 + k] + q[d * 3 + k];        // pos_j - pos_i + delta_i
      } else if (k < 303) {
        v = h[(size_t)s * DF + (k - 3)];          // x_j (L2-resident gather)
      }
    }
    As[idx] = v;
  }
  if (tid < 16) {
    int e = e0 + tid;
    dsts[tid] = (e < E) ? dst[e] : -1;
  }
  __syncthreads();

  const int lane = tid & 31;
  const int wave = tid >> 5;
  const int ln   = lane & 15;
  const int hi   = lane >> 4;
  const int koff = hi << 1;
  const int mofs = hi << 3;
  const long bstride = (long)KPAD * 2;    // NPAD==KPAD==304 for the edge MLP

  for (int ntile = wave; ntile < 19; ntile += 4) {
    const int n = ntile * 16 + ln;
    const v2f* ap = (const v2f*)(As + ln * KPAD + koff);
    const v2f* bp = fWp + ((long)n * 2 + hi);
    v8f acc = {};
#pragma unroll 4
    for (int c = 0; c < 76; ++c) {        // 76 chunks = K 304
      v2f a = ap[2 * c];
      v2f b = bp[(long)c * bstride];
      acc = __builtin_amdgcn_wmma_f32_16x16x4_f32(false, a, false, b,
                                                  (short)0, acc, false, false);
    }
    if (n < DF) {
      const float bv = fb[n];
#pragma unroll
      for (int r = 0; r < 8; ++r) {
        const int d = dsts[mofs + r];
        if (d >= 0) atomicMaxF(&agg[(size_t)d * DF + n], acc[r] + bv);
      }
    }
  }
}

// ---------------------------------------------------------------------------
// q[n] = h[n] @ hW + hb - pos[n]   (so edge rel = pos[src] + q[dst])
// ---------------------------------------------------------------------------
__global__ __launch_bounds__(128)
void qdelta_kernel(const float* __restrict__ h, const float* __restrict__ pos,
                   const float* __restrict__ hW, const float* __restrict__ hb,
                   float* __restrict__ q, int Nn)
{
  int n = blockIdx.x * blockDim.x + threadIdx.x;
  if (n >= Nn) return;
  float a0 = hb[0], a1 = hb[1], a2 = hb[2];
  const float* hr = h + (size_t)n * DF;
  for (int k = 0; k < DF; ++k) {
    float v = hr[k];
    a0 += v * hW[k * 3 + 0];
    a1 += v * hW[k * 3 + 1];
    a2 += v * hW[k * 3 + 2];
  }
  q[n * 3 + 0] = a0 - pos[n * 3 + 0];
  q[n * 3 + 1] = a1 - pos[n * 3 + 1];
  q[n * 3 + 2] = a2 - pos[n * 3 + 2];
}

__global__ void fill_u32(unsigned int* __restrict__ p, unsigned int val, long n)
{
  long i = (long)blockIdx.x * blockDim.x + threadIdx.x;
  if (i < n) p[i] = val;
}

// one block per node; 320 threads cover 300 features
__global__ __launch_bounds__(320)
void pool_kernel(const float* __restrict__ h, const int* __restrict__ batch,
                 float* __restrict__ sums, float* __restrict__ cnt)
{
  int n = blockIdx.x;
  int f = threadIdx.x;
  int g = batch[n];
  if (f < DF) atomicAdd(&sums[(size_t)g * DF + f], h[(size_t)n * DF + f]);
  if (f == 0) atomicAdd(&cnt[g], 1.0f);
}

// one wave (32 lanes) per graph: mean -> logits -> log_softmax
__global__ __launch_bounds__(1024)
void head_kernel(const float* __restrict__ sums, const float* __restrict__ cnt,
                 const float* __restrict__ dW, const float* __restrict__ db,
                 float* __restrict__ out)
{
  const int g    = threadIdx.x >> 5;
  const int lane = threadIdx.x & 31;
  float c = cnt[g];
  c = c > 1.0f ? c : 1.0f;
  float logit[2];
#pragma unroll
  for (int t = 0; t < 2; ++t) {
    int cl = lane + t * 32;
    logit[t] = -INFINITY;
    if (cl < NCLS) {
      float acc = db[cl];
      for (int k = 0; k < DF; ++k)
        acc += (sums[(size_t)g * DF + k] / c) * dW[k * NCLS + cl];
      logit[t] = acc;
    }
  }
  float mx = fmaxf(logit[0], logit[1]);
  for (int o = 16; o > 0; o >>= 1) mx = fmaxf(mx, __shfl_xor(mx, o, 32));
  float s = 0.0f;
#pragma unroll
  for (int t = 0; t < 2; ++t) {
    int cl = lane + t * 32;
    if (cl < NCLS) s += __expf(logit[t] - mx);
  }
  for (int o = 16; o > 0; o >>= 1) s += __shfl_xor(s, o, 32);
  float lse = mx + __logf(s);
#pragma unroll
  for (int t = 0; t < 2; ++t) {
    int cl = lane + t * 32;
    if (cl < NCLS) out[g * NCLS + cl] = logit[t] - lse;
  }
}

// ---------------------------------------------------------------------------
static inline long packed_elems(int K, int NPAD) {
  int KC = (K + 3) / 4;
  return (long)KC * NPAD * 4;
}

extern "C" void kernel_launch(void* const* d_in, const int* in_sizes, int n_in,
                              void* d_out, int out_size, void* d_ws, size_t ws_size,
                              hipStream_t stream)
{
  const float* x     = (const float*)d_in[0];     // [N,3] (also 'pos')
  const int*   ei    = (const int*)  d_in[1];     // [2,E]
  const int*   batch = (const int*)  d_in[2];     // [N]
  const float* pW1 = (const float*)d_in[3];  const float* pb1 = (const float*)d_in[4];
  const float* pW2 = (const float*)d_in[5];  const float* pb2 = (const float*)d_in[6];
  const float* pW3 = (const float*)d_in[7];  const float* pb3 = (const float*)d_in[8];
  const float* hW  = (const float*)d_in[9];  const float* hb  = (const float*)d_in[10];
  const float* fW  = (const float*)d_in[11]; const float* fb  = (const float*)d_in[12];
  const float* gW  = (const float*)d_in[13]; const float* gb  = (const float*)d_in[14];
  const float* lW  = (const float*)d_in[15]; const float* lb  = (const float*)d_in[16];
  const float* dW  = (const float*)d_in[17]; const float* db  = (const float*)d_in[18];

  const int Nn = NNODES, E = NEDGES;
  const int* src  = ei;
  const int* dstA = ei + E;

  // ---- workspace layout (floats) ----
  float* h    = (float*)d_ws;                       // [N,300]
  float* tmp  = h    + (size_t)Nn * DF;             // [N,300] (also t1)
  float* agg  = tmp  + (size_t)Nn * DF;             // [N,300] (also t2)
  float* q    = agg  + (size_t)Nn * DF;             // [N,3]
  float* sums = q    + (size_t)Nn * 3;              // [32,300]
  float* cnt  = sums + (size_t)NGRAPH * DF;         // [32]
  float* pk   = cnt  + NGRAPH;                      // packed weights follow

  float* pk1 = pk;                                   // K=3,   NPAD=64
  float* pk2 = pk1 + packed_elems(3, 64);            // K=64,  NPAD=128
  float* pk3 = pk2 + packed_elems(64, 128);          // K=128, NPAD=304
  float* pfW[3]; float* pgW[3]; float* plW[3];
  float* cur = pk3 + packed_elems(128, 304);
  for (int i = 0; i < 3; ++i) { pfW[i] = cur; cur += packed_elems(303, 304); }
  for (int i = 0; i < 3; ++i) { pgW[i] = cur; cur += packed_elems(300, 304); }
  for (int i = 0; i < 3; ++i) { plW[i] = cur; cur += packed_elems(300, 304); }

  // ---- pack all weight matrices (cheap; ~3.5 MB total) ----
  {
    long n1 = packed_elems(3, 64);
    pack_b<<<(int)((n1 + 255) / 256), 256, 0, stream>>>(pW1, pk1, 3, 64, 64, 1);
    long n2 = packed_elems(64, 128);
    pack_b<<<(int)((n2 + 255) / 256), 256, 0, stream>>>(pW2, pk2, 64, 128, 128, 16);
    long n3 = packed_elems(128, 304);
    pack_b<<<(int)((n3 + 255) / 256), 256, 0, stream>>>(pW3, pk3, 128, DF, 304, 32);
    long nf = packed_elems(303, 304);
    long ng = packed_elems(300, 304);
    for (int i = 0; i < 3; ++i) {
      pack_b<<<(int)((nf + 255) / 256), 256, 0, stream>>>(
          fW + (size_t)i * 303 * DF, pfW[i], 303, DF, 304, 76);
      pack_b<<<(int)((ng + 255) / 256), 256, 0, stream>>>(
          gW + (size_t)i * DF * DF, pgW[i], DF, DF, 304, 75);
      pack_b<<<(int)((ng + 255) / 256), 256, 0, stream>>>(
          lW + (size_t)i * DF * DF, plW[i], DF, DF, 304, 75);
    }
  }

  const int mt = (Nn + 15) / 16;                    // m-tiles over nodes

  // embed MLP: 3 -> 64 -> 128 -> 300 (relu, relu, linear)
  gemm16_wmma<<<mt, 128, 0, stream>>>(x,   (const v2f*)pk1, pb1, nullptr, tmp,
                                      Nn, 64, 3, 64, 1, 1);
  gemm16_wmma<<<mt, 128, 0, stream>>>(tmp, (const v2f*)pk2, pb2, nullptr, agg,
                                      Nn, 128, 64, 128, 16, 1);
  gemm16_wmma<<<mt, 128, 0, stream>>>(agg, (const v2f*)pk3, pb3, nullptr, h,
                                      Nn, DF, 128, 304, 32, 0);

  const long aggN = (long)Nn * DF;
  for (int i = 0; i < 3; ++i) {
    qdelta_kernel<<<(Nn + 127) / 128, 128, 0, stream>>>(
        h, x, hW + (size_t)i * DF * 3, hb + i * 3, q, Nn);
    fill_u32<<<(int)((aggN + 255) / 256), 256, 0, stream>>>(
        (unsigned int*)agg, 0xFF800000u /* -inf */, aggN);
    edge_wmma<<<(E + 15) / 16, 128, 0, stream>>>(
        h, x, q, src, dstA, (const v2f*)pfW[i], fb + (size_t)i * DF, agg, E);
    // relu(fixinf(agg) @ gW + gb + h) -> tmp
    gemm16_wmma<<<mt, 128, 0, stream>>>(
        agg, (const v2f*)pgW[i], gb + (size_t)i * DF, h, tmp,
        Nn, DF, DF, 304, 75, 1 | 2);
    // tmp @ lW + lb -> h
    gemm16_wmma<<<mt, 128, 0, stream>>>(
        tmp, (const v2f*)plW[i], lb + (size_t)i * DF, nullptr, h,
        Nn, DF, DF, 304, 75, 0);
  }

  // global mean pool + head
  const long poolN = (long)NGRAPH * DF + NGRAPH;    // sums + cnt contiguous
  fill_u32<<<(int)((poolN + 255) / 256), 256, 0, stream>>>((unsigned int*)sums, 0u, poolN);
  pool_kernel<<<Nn, 320, 0, stream>>>(h, batch, sums, cnt);
  head_kernel<<<1, 1024, 0, stream>>>(sums, cnt, dW, db, (float*)d_out);
}